// CSAttention_14224931684888
// MI455X (gfx1250) — compile-verified
//
#include <hip/hip_runtime.h>

#define C_DIM 256
#define B_DIM 4
#define HIMG_ 128
#define NTOK 16384
#define P_DIM 32
#define WS_ 4
#define HEAD_ 8
#define HD_ 32

typedef float v2f __attribute__((ext_vector_type(2)));
typedef float v8f __attribute__((ext_vector_type(8)));

__device__ __forceinline__ v8f wmma4(v2f a, v2f b, v8f c) {
  // D = A(16x4 f32) * B(4x16 f32) + C(16x16 f32)
  return __builtin_amdgcn_wmma_f32_16x16x4_f32(false, a, false, b, (short)0, c,
                                               false, false);
}

__device__ __forceinline__ v8f zero8() {
  v8f z;
#pragma unroll
  for (int i = 0; i < 8; ++i) z[i] = 0.0f;
  return z;
}

__device__ __forceinline__ v2f ldb(const float* __restrict__ W, int ka, int nn) {
  v2f b;
  b.x = W[(size_t)ka * C_DIM + nn];
  b.y = W[(size_t)(ka + 1) * C_DIM + nn];
  return b;
}

// ---------------------------------------------------------------------------
// A-tile loader shared by the GEMM kernels. mode==1: rows gathered in window
// order from the (B,N,C) image tensor; mode==0: linear rows.
// ---------------------------------------------------------------------------
__device__ __forceinline__ void load_a_tile(float (*a_s)[260],
                                            const float* __restrict__ X,
                                            int mode, int blk, int t) {
  int r = t >> 4;
  int c0 = t & 15;
  const float* src;
  if (mode == 1) {
    int win = blk;  // tile == one 4x4 window
    int b = win / (P_DIM * P_DIM);
    int pq = win % (P_DIM * P_DIM);
    int p = pq / P_DIM, q = pq % P_DIM;
    int y = p * WS_ + (r >> 2);
    int x = q * WS_ + (r & 3);
    src = X + ((size_t)b * NTOK + (size_t)y * HIMG_ + x) * C_DIM;
  } else {
    src = X + (size_t)(blk * 16 + r) * C_DIM;
  }
#pragma unroll
  for (int j = 0; j < 16; ++j) a_s[r][c0 + 16 * j] = src[c0 + 16 * j];
}

// ---------------------------------------------------------------------------
// GEMM, 3 outputs sharing one A tile (branch-free inner loop).
// One block = one 16-row tile; 8 waves cover 16 col-tiles (2 each) per output.
// ---------------------------------------------------------------------------
__global__ __launch_bounds__(256) void gemm3_k(
    const float* __restrict__ X, int mode,
    const float* __restrict__ W0, const float* __restrict__ Bi0, float* __restrict__ O0,
    const float* __restrict__ W1, const float* __restrict__ Bi1, float* __restrict__ O1,
    const float* __restrict__ W2, const float* __restrict__ Bi2, float* __restrict__ O2) {
  __shared__ float a_s[16][260];  // 16 rows x 256 cols, +4 pad (even stride)
  const int m0 = blockIdx.x * 16;
  const int t = threadIdx.x;

  load_a_tile(a_s, X, mode, blockIdx.x, t);

  // warm L2/WGP$ for the weight columns this block will stream
  __builtin_prefetch(&W0[t * C_DIM], 0, 1);
  __builtin_prefetch(&W1[t * C_DIM], 0, 1);
  __builtin_prefetch(&W2[t * C_DIM], 0, 1);

  __syncthreads();

  const int wv = t >> 5;
  const int lane = t & 31;
  const int l = lane & 15;
  const int half = lane >> 4;
  const int n0 = wv * 32;
  const int nn0 = n0 + l;
  const int nn1 = n0 + 16 + l;

  v8f acc[3][2];
#pragma unroll
  for (int o = 0; o < 3; ++o)
#pragma unroll
    for (int j = 0; j < 2; ++j) acc[o][j] = zero8();

#pragma unroll 2
  for (int k0 = 0; k0 < C_DIM; k0 += 4) {
    const int ka = k0 + 2 * half;
    v2f a = *(const v2f*)&a_s[l][ka];
    v2f b00 = ldb(W0, ka, nn0), b01 = ldb(W0, ka, nn1);
    v2f b10 = ldb(W1, ka, nn0), b11 = ldb(W1, ka, nn1);
    v2f b20 = ldb(W2, ka, nn0), b21 = ldb(W2, ka, nn1);
    acc[0][0] = wmma4(a, b00, acc[0][0]);
    acc[0][1] = wmma4(a, b01, acc[0][1]);
    acc[1][0] = wmma4(a, b10, acc[1][0]);
    acc[1][1] = wmma4(a, b11, acc[1][1]);
    acc[2][0] = wmma4(a, b20, acc[2][0]);
    acc[2][1] = wmma4(a, b21, acc[2][1]);
  }

#pragma unroll
  for (int j = 0; j < 2; ++j) {
    const int nn = n0 + 16 * j + l;
    const float b0 = Bi0[nn], b1 = Bi1[nn], b2 = Bi2[nn];
#pragma unroll
    for (int r = 0; r < 8; ++r) {
      const size_t row = (size_t)(m0 + r + 8 * half) * C_DIM + nn;
      O0[row] = acc[0][j][r] + b0;
      O1[row] = acc[1][j][r] + b1;
      O2[row] = acc[2][j][r] + b2;
    }
  }
}

// ---------------------------------------------------------------------------
// GEMM, single output (branch-free).
// ---------------------------------------------------------------------------
__global__ __launch_bounds__(256) void gemm1_k(const float* __restrict__ X,
                                               int mode,
                                               const float* __restrict__ W0,
                                               const float* __restrict__ Bi0,
                                               float* __restrict__ O0) {
  __shared__ float a_s[16][260];
  const int m0 = blockIdx.x * 16;
  const int t = threadIdx.x;

  load_a_tile(a_s, X, mode, blockIdx.x, t);
  __builtin_prefetch(&W0[t * C_DIM], 0, 1);
  __syncthreads();

  const int wv = t >> 5;
  const int lane = t & 31;
  const int l = lane & 15;
  const int half = lane >> 4;
  const int n0 = wv * 32;
  const int nn0 = n0 + l;
  const int nn1 = n0 + 16 + l;

  v8f acc0 = zero8(), acc1 = zero8();
#pragma unroll 4
  for (int k0 = 0; k0 < C_DIM; k0 += 4) {
    const int ka = k0 + 2 * half;
    v2f a = *(const v2f*)&a_s[l][ka];
    v2f b0 = ldb(W0, ka, nn0);
    v2f b1 = ldb(W0, ka, nn1);
    acc0 = wmma4(a, b0, acc0);
    acc1 = wmma4(a, b1, acc1);
  }

  const float bb0 = Bi0[nn0], bb1 = Bi0[nn1];
#pragma unroll
  for (int r = 0; r < 8; ++r) {
    const size_t row = (size_t)(m0 + r + 8 * half) * C_DIM;
    O0[row + nn0] = acc0[r] + bb0;
    O0[row + nn1] = acc1[r] + bb1;
  }
}

// ---------------------------------------------------------------------------
// Window attention: one block per 4x4 window, one wave per head.
// scores = Q Kᵀ / sqrt(HD) -> softmax (LDS) -> ctx = P V   (all WMMA f32)
// ---------------------------------------------------------------------------
__global__ __launch_bounds__(256) void winattn_k(const float* __restrict__ Q,
                                                 const float* __restrict__ K,
                                                 const float* __restrict__ V,
                                                 float* __restrict__ CTX) {
  __shared__ float p_s[HEAD_][16][18];  // padded, even stride
  const int win = blockIdx.x;
  const size_t base = (size_t)win * 16 * C_DIM;
  const int t = threadIdx.x;
  const int h = t >> 5;
  const int lane = t & 31;
  const int l = lane & 15, half = lane >> 4;
  const float* Qh = Q + base + h * HD_;
  const float* Kh = K + base + h * HD_;
  const float* Vh = V + base + h * HD_;

  v8f s = zero8();
#pragma unroll
  for (int k0 = 0; k0 < HD_; k0 += 4) {
    const int ka = k0 + 2 * half;
    v2f a = *(const v2f*)&Qh[(size_t)l * C_DIM + ka];
    v2f b = *(const v2f*)&Kh[(size_t)l * C_DIM + ka];  // B[k][n=j] = K[j][k]
    s = wmma4(a, b, s);
  }
  const float scale = 0.17677669529663687f;  // 1/sqrt(32)
#pragma unroll
  for (int r = 0; r < 8; ++r) p_s[h][r + 8 * half][l] = s[r] * scale;
  __syncthreads();

  if (t < HEAD_ * 16) {  // row-wise softmax over 16 keys
    float* pr = p_s[t >> 4][t & 15];
    float mx = pr[0];
    for (int j = 1; j < 16; ++j) mx = fmaxf(mx, pr[j]);
    float sum = 0.f;
    for (int j = 0; j < 16; ++j) { float e = expf(pr[j] - mx); pr[j] = e; sum += e; }
    float inv = 1.f / sum;
    for (int j = 0; j < 16; ++j) pr[j] *= inv;
  }
  __syncthreads();

#pragma unroll
  for (int j = 0; j < 2; ++j) {  // ctx: 16x32 per head = two 16x16 tiles
    const int n0 = 16 * j;
    v8f acc = zero8();
#pragma unroll
    for (int k0 = 0; k0 < 16; k0 += 4) {
      const int ka = k0 + 2 * half;
      v2f a;
      a.x = p_s[h][l][ka];
      a.y = p_s[h][l][ka + 1];
      v2f b;
      b.x = Vh[(size_t)ka * C_DIM + n0 + l];
      b.y = Vh[(size_t)(ka + 1) * C_DIM + n0 + l];
      acc = wmma4(a, b, acc);
    }
#pragma unroll
    for (int r = 0; r < 8; ++r)
      CTX[base + (size_t)(r + 8 * half) * C_DIM + h * HD_ + n0 + l] = acc[r];
  }
}

// ---------------------------------------------------------------------------
// DlightConv: per-window mean -> 256->16 proj -> softmax -> weighted pool
// ---------------------------------------------------------------------------
__global__ __launch_bounds__(256) void dlight_k(const float* __restrict__ WOUT,
                                                const float* __restrict__ WD,
                                                const float* __restrict__ BD,
                                                float* __restrict__ XD) {
  __shared__ float tile[16][C_DIM];
  __shared__ float avg_s[C_DIM];
  __shared__ float logit_s[16];
  __shared__ float prob_s[16];
  const int win = blockIdx.x;
  const int t = threadIdx.x;  // channel
  const float* src = WOUT + (size_t)win * 16 * C_DIM;
#pragma unroll
  for (int w = 0; w < 16; ++w) tile[w][t] = src[(size_t)w * C_DIM + t];
  float s = 0.f;
#pragma unroll
  for (int w = 0; w < 16; ++w) s += tile[w][t];
  avg_s[t] = s * (1.0f / 16.0f);
  __syncthreads();
  if (t < 16) {
    float acc = BD[t];
    for (int c = 0; c < C_DIM; ++c) acc += avg_s[c] * WD[c * 16 + t];
    logit_s[t] = acc;
  }
  __syncthreads();
  if (t == 0) {
    float mx = logit_s[0];
    for (int j = 1; j < 16; ++j) mx = fmaxf(mx, logit_s[j]);
    float sum = 0.f;
    for (int j = 0; j < 16; ++j) { float e = expf(logit_s[j] - mx); prob_s[j] = e; sum += e; }
    float inv = 1.f / sum;
    for (int j = 0; j < 16; ++j) prob_s[j] *= inv;
  }
  __syncthreads();
  float o = 0.f;
#pragma unroll
  for (int w = 0; w < 16; ++w) o += tile[w][t] * prob_s[w];
  XD[(size_t)win * C_DIM + t] = o;
}

// ---------------------------------------------------------------------------
// Axial attention + GaussianTrans. mode 0: per-(b,row), stride C; writes nv.
// mode 1: per-(b,col), stride P*C; accumulates into nv.
// scores 32x32 via WMMA (K=256) -> biased softmax in LDS -> S@V via WMMA.
// ---------------------------------------------------------------------------
__global__ __launch_bounds__(128) void axial_k(
    const float* __restrict__ Q2, const float* __restrict__ K2,
    const float* __restrict__ V2, float* __restrict__ NV,
    const float* __restrict__ GS, const float* __restrict__ GB, int mode) {
  __shared__ float s_s[32][33];
  const int b = blockIdx.x / P_DIM;
  const int idx = blockIdx.x % P_DIM;
  size_t base;
  int rs;
  if (mode == 0) {
    base = ((size_t)b * P_DIM + idx) * P_DIM * C_DIM;  // fixed row r
    rs = C_DIM;
  } else {
    base = ((size_t)b * P_DIM * P_DIM + idx) * C_DIM;  // fixed col c
    rs = P_DIM * C_DIM;
  }
  const float* Qb = Q2 + base;
  const float* Kb = K2 + base;
  const float* Vb = V2 + base;
  float* Ob = NV + base;
  const int t = threadIdx.x;
  const int wv = t >> 5, lane = t & 31, l = lane & 15, half = lane >> 4;

  {  // scores: 4 waves x one 16x16 tile, full K=256
    const int m0 = (wv & 1) * 16, n0 = (wv >> 1) * 16;
    v8f acc = zero8();
#pragma unroll 2
    for (int k0 = 0; k0 < C_DIM; k0 += 4) {
      const int ka = k0 + 2 * half;
      v2f a = *(const v2f*)&Qb[(size_t)(m0 + l) * rs + ka];
      v2f bb = *(const v2f*)&Kb[(size_t)(n0 + l) * rs + ka];
      acc = wmma4(a, bb, acc);
    }
#pragma unroll
    for (int r = 0; r < 8; ++r) s_s[m0 + r + 8 * half][n0 + l] = acc[r];
  }
  __syncthreads();

  if (t < 32) {  // Gaussian bias + softmax over keys
    const float gs = GS[0], gb = GB[0];
    float* row = s_s[t];
    float mx = -1e30f;
    for (int n = 0; n < 32; ++n) {
      float d = (float)(t - n);
      float vv = row[n] - (gs * d * d + gb);
      row[n] = vv;
      mx = fmaxf(mx, vv);
    }
    float sum = 0.f;
    for (int n = 0; n < 32; ++n) { float e = expf(row[n] - mx); row[n] = e; sum += e; }
    float inv = 1.f / sum;
    for (int n = 0; n < 32; ++n) row[n] *= inv;
  }
  __syncthreads();

  for (int tt = 0; tt < 8; ++tt) {  // out = S(32x32) @ V(32x256)
    const int tileIdx = wv * 8 + tt;
    const int m0 = (tileIdx / 16) * 16;
    const int n0 = (tileIdx % 16) * 16;
    v8f acc = zero8();
#pragma unroll
    for (int k0 = 0; k0 < 32; k0 += 4) {
      const int ka = k0 + 2 * half;
      v2f a;
      a.x = s_s[m0 + l][ka];
      a.y = s_s[m0 + l][ka + 1];
      v2f bb;
      bb.x = Vb[(size_t)ka * rs + n0 + l];
      bb.y = Vb[(size_t)(ka + 1) * rs + n0 + l];
      acc = wmma4(a, bb, acc);
    }
#pragma unroll
    for (int r = 0; r < 8; ++r) {
      const size_t o = (size_t)(m0 + r + 8 * half) * rs + n0 + l;
      if (mode == 0)
        Ob[o] = acc[r];
      else
        Ob[o] += acc[r];
    }
  }
}

// ---------------------------------------------------------------------------
// Final: A row (512) = [bilinear-upsampled nv | un-windowed win_out],
// out = A @ w_conv (512x256) + b_conv. One block per 16 consecutive pixels.
// ---------------------------------------------------------------------------
__global__ __launch_bounds__(256) void final_k(const float* __restrict__ NV,
                                               const float* __restrict__ WOUT,
                                               const float* __restrict__ WC,
                                               const float* __restrict__ BC,
                                               float* __restrict__ OUT) {
  __shared__ float a_s[16][516];  // 512 + pad 4 (even stride)
  const int tileId = blockIdx.x;
  const int b = tileId >> 10;  // 1024 tiles per image
  const int r2 = tileId & 1023;
  const int y = r2 >> 3;
  const int x0 = (r2 & 7) * 16;
  const int t = threadIdx.x;

  const float syf = (float)y * (31.0f / 127.0f);  // align_corners 32->128
  const int y0 = (int)floorf(syf);
  const int y1 = min(y0 + 1, 31);
  const float ty = syf - (float)y0;

  __builtin_prefetch(&WC[t * C_DIM], 0, 1);
  __builtin_prefetch(&WC[(C_DIM + t) * C_DIM], 0, 1);

  {
    const int c = t;  // channel
    const size_t nb = (size_t)b * P_DIM * P_DIM * C_DIM;
    for (int r = 0; r < 16; ++r) {
      const int x = x0 + r;
      const float sxf = (float)x * (31.0f / 127.0f);
      const int xx0 = (int)floorf(sxf);
      const int xx1 = min(xx0 + 1, 31);
      const float tx = sxf - (float)xx0;
      const float n00 = NV[nb + ((size_t)y0 * P_DIM + xx0) * C_DIM + c];
      const float n01 = NV[nb + ((size_t)y0 * P_DIM + xx1) * C_DIM + c];
      const float n10 = NV[nb + ((size_t)y1 * P_DIM + xx0) * C_DIM + c];
      const float n11 = NV[nb + ((size_t)y1 * P_DIM + xx1) * C_DIM + c];
      a_s[r][c] = (1.f - ty) * ((1.f - tx) * n00 + tx * n01) +
                  ty * ((1.f - tx) * n10 + tx * n11);
      const int p = y >> 2, q = x >> 2;
      const int w = (y & 3) * 4 + (x & 3);
      a_s[r][C_DIM + c] =
          WOUT[((((size_t)b * P_DIM + p) * P_DIM + q) * 16 + w) * C_DIM + c];
    }
  }
  __syncthreads();

  const int wv = t >> 5, lane = t & 31, l = lane & 15, half = lane >> 4;
  const int n0b = wv * 32;
  const int nn0 = n0b + l;
  const int nn1 = n0b + 16 + l;
  v8f acc0 = zero8(), acc1 = zero8();
#pragma unroll 2
  for (int k0 = 0; k0 < 2 * C_DIM; k0 += 4) {
    const int ka = k0 + 2 * half;
    v2f a = *(const v2f*)&a_s[l][ka];
    v2f b0 = ldb(WC, ka, nn0);
    v2f b1 = ldb(WC, ka, nn1);
    acc0 = wmma4(a, b0, acc0);
    acc1 = wmma4(a, b1, acc1);
  }
  const size_t rowbase = (size_t)b * NTOK + (size_t)y * HIMG_ + x0;
  const float bias0 = BC[nn0], bias1 = BC[nn1];
#pragma unroll
  for (int r = 0; r < 8; ++r) {
    const size_t row = (rowbase + r + 8 * half) * C_DIM;
    OUT[row + nn0] = acc0[r] + bias0;
    OUT[row + nn1] = acc1[r] + bias1;
  }
}

// ---------------------------------------------------------------------------
extern "C" void kernel_launch(void* const* d_in, const int* in_sizes, int n_in,
                              void* d_out, int out_size, void* d_ws,
                              size_t ws_size, hipStream_t stream) {
  const float* x = (const float*)d_in[0];
  const float* wq1 = (const float*)d_in[1];
  const float* bq1 = (const float*)d_in[2];
  const float* wk1 = (const float*)d_in[3];
  const float* bk1 = (const float*)d_in[4];
  const float* wv1 = (const float*)d_in[5];
  const float* bv1 = (const float*)d_in[6];
  const float* wo1 = (const float*)d_in[7];
  const float* bo1 = (const float*)d_in[8];
  const float* wd = (const float*)d_in[9];
  const float* bd = (const float*)d_in[10];
  const float* wq2 = (const float*)d_in[11];
  const float* bq2 = (const float*)d_in[12];
  const float* wk2 = (const float*)d_in[13];
  const float* bk2 = (const float*)d_in[14];
  const float* wv2 = (const float*)d_in[15];
  const float* bv2 = (const float*)d_in[16];
  const float* gs = (const float*)d_in[17];
  const float* gb = (const float*)d_in[18];
  const float* w_conv = (const float*)d_in[19];
  const float* b_conv = (const float*)d_in[20];
  float* out = (float*)d_out;

  const size_t T = (size_t)B_DIM * NTOK * C_DIM;           // 16.78M floats
  const size_t S = (size_t)B_DIM * P_DIM * P_DIM * C_DIM;  // 1.05M floats
  float* ws = (float*)d_ws;
  float* q1 = ws;
  float* k1 = ws + T;
  float* v1 = ws + 2 * T;
  float* ctx = ws + 3 * T;
  float* win_out = q1;  // q1 dead after winattn
  float* xd = k1;       // k1 region dead after winattn
  float* q2 = k1 + S;
  float* k2 = k1 + 2 * S;
  float* v2 = k1 + 3 * S;
  float* nv = k1 + 4 * S;

  const int MWIN = B_DIM * P_DIM * P_DIM * 16;  // 65536 windowed rows
  const int NWIN = B_DIM * P_DIM * P_DIM;       // 4096 windows

  // 1) q/k/v projections (windowed gather)
  gemm3_k<<<MWIN / 16, 256, 0, stream>>>(x, 1, wq1, bq1, q1, wk1, bk1, k1, wv1,
                                         bv1, v1);
  // 2) window attention
  winattn_k<<<NWIN, 256, 0, stream>>>(q1, k1, v1, ctx);
  // 3) output projection -> win_out (reuses q1 space)
  gemm1_k<<<MWIN / 16, 256, 0, stream>>>(ctx, 0, wo1, bo1, win_out);
  // 4) DlightConv pooling
  dlight_k<<<NWIN, 256, 0, stream>>>(win_out, wd, bd, xd);
  // 5) axial q2/k2/v2 projections
  gemm3_k<<<(int)(S / C_DIM) / 16, 256, 0, stream>>>(xd, 0, wq2, bq2, q2, wk2,
                                                     bk2, k2, wv2, bv2, v2);
  // 6) axial attention: rows then columns (accumulate)
  axial_k<<<B_DIM * P_DIM, 128, 0, stream>>>(q2, k2, v2, nv, gs, gb, 0);
  axial_k<<<B_DIM * P_DIM, 128, 0, stream>>>(q2, k2, v2, nv, gs, gb, 1);
  // 7) upsample + concat + 1x1 conv
  final_k<<<B_DIM * 1024, 256, 0, stream>>>(nv, win_out, w_conv, b_conv, out);
}